// LINKX_59133109731532
// MI455X (gfx1250) — compile-verified
//
#include <hip/hip_runtime.h>
#include <hip/hip_bf16.h>

typedef float v8f __attribute__((ext_vector_type(8)));
typedef float v2f __attribute__((ext_vector_type(2)));

#define Nn   100000
#define Ee   1600000
#define Hh   128
#define OUTC 64
#define WDIM 512
#define RNK  10

__device__ __forceinline__ float leaky(float v) { return v >= 0.f ? v : 0.01f * v; }

// ---------------- workspace zero-init ----------------
__global__ void zero_ws(float* __restrict__ p, int n) {
  int i = blockIdx.x * blockDim.x + threadIdx.x;
  int stride = gridDim.x * blockDim.x;
  for (; i < n; i += stride) p[i] = 0.f;
}

// ---------------- styles = w_row @ aff_w + aff_b ----------------
__global__ void styles_kernel(const float* __restrict__ wv, const float* __restrict__ aff_w,
                              const float* __restrict__ aff_b, float* __restrict__ styles,
                              int ncols) {
  int j = blockIdx.x * blockDim.x + threadIdx.x;
  if (j >= ncols) return;
  float s = aff_b[j];
  for (int k = 0; k < WDIM; ++k) s = fmaf(wv[k], aff_w[k * ncols + j], s);
  styles[j] = s;
}

// ------- modulated, row-normalized weight, stored transposed [c_in][c_out] -------
__global__ void modw_kernel(const float* __restrict__ weight, const float* __restrict__ styles,
                            float* __restrict__ Wt, int c_out, int c_in) {
  int i = blockIdx.x;   // output row (c_out)
  int j = threadIdx.x;  // input col (c_in); blockDim.x == c_in == 128
  const float* left  = styles;                // [c_out][RNK]
  const float* right = styles + c_out * RNK;  // [RNK][c_in]
  float mod = 0.f;
#pragma unroll
  for (int r = 0; r < RNK; ++r) mod = fmaf(left[i * RNK + r], right[r * c_in + j], mod);
  mod *= 0.31622776601683794f;  // 1/sqrt(RANK)
  float wm = weight[i * c_in + j] * (mod + 1.f);
  __shared__ float red[128];
  red[j] = wm * wm;
  __syncthreads();
  for (int s = 64; s > 0; s >>= 1) {
    if (j < s) red[j] += red[j + s];
    __syncthreads();
  }
  float inv = 1.f / (sqrtf(red[0]) + 1e-8f);
  Wt[j * c_out + i] = wm * inv;
}

// ---------------- edge gather + scatter-add (L2-resident) ----------------
__global__ void edge_kernel(const float* __restrict__ elw, const int* __restrict__ ei,
                            const float* __restrict__ ew, float* __restrict__ acc, int E) {
  int e = blockIdx.x * 2 + (threadIdx.x >> 7);  // 2 edges per 256-thread block
  int c = threadIdx.x & 127;                    // channel
  if (e >= E) return;
  int src = ei[e];       // edge_index[0][e]
  int dst = ei[E + e];   // edge_index[1][e]
  float wv = ew[e];
  atomicAdd(&acc[dst * 128 + c], elw[src * 128 + c] * wv);
}

// ---------------- fp32 WMMA tile GEMM: C += A(16x128, LDS) * B(128xldb, global) ----------------
__device__ __forceinline__ v8f wmma_k128(const float* __restrict__ A,  // LDS, row-major [16][128]
                                         const float* __restrict__ B,  // global [128][ldb]
                                         int ldb, int col0, v8f acc) {
  const int lane = threadIdx.x & 31;
  const int n  = lane & 15;          // A row / B,C column within tile
  const int kh = (lane >> 4) << 1;   // 0 for lanes 0-15, 2 for lanes 16-31
#pragma unroll 8
  for (int k0 = 0; k0 < 128; k0 += 4) {
    v2f a = *(const v2f*)(A + n * 128 + k0 + kh);  // A[m][k0+kh], A[m][k0+kh+1]
    v2f b;
    b.x = B[(k0 + kh) * ldb + col0 + n];
    b.y = B[(k0 + kh + 1) * ldb + col0 + n];
    acc = __builtin_amdgcn_wmma_f32_16x16x4_f32(false, a, false, b, (short)0, acc, false, false);
  }
  return acc;
}

// ---------------- fused dense chain: 16 nodes per block, 8 waves = 8 column tiles ----------------
__global__ void __launch_bounds__(256) fused_dense(
    const float* __restrict__ out_acc, const float* __restrict__ elb,
    const float* __restrict__ xg,
    const float* __restrict__ node_w, const float* __restrict__ node_b,
    const float* __restrict__ cat1_w, const float* __restrict__ cat1_b,
    const float* __restrict__ cat2_w, const float* __restrict__ cat2_b,
    const float* __restrict__ W0t, const float* __restrict__ b0,
    const float* __restrict__ W1t, const float* __restrict__ b1,
    float* __restrict__ outp) {
  __shared__ float tA[16 * 128];  // aggregated + edge bias
  __shared__ float xA[16 * 128];  // x tile
  __shared__ float xn[16 * 128];  // x @ node_w + node_b
  __shared__ float h1[16 * 128];
  __shared__ float h2[16 * 128];

  const int tid = threadIdx.x;
  const int tile0 = blockIdx.x * 16;

  for (int idx = tid; idx < 16 * 128; idx += 256) {
    int r = idx >> 7, c = idx & 127;
    int node = tile0 + r;
    tA[idx] = out_acc[node * 128 + c] + elb[c];
    xA[idx] = xg[node * 128 + c];
  }
  __syncthreads();

  const int wave = tid >> 5;
  const int lane = tid & 31;
  const int n = lane & 15;
  const int khsel = lane >> 4;   // C-tile row group
  const int col0 = wave * 16;

  // stage 1a: xn = x @ node_w + node_b
  {
    v8f acc = {};
    acc = wmma_k128(xA, node_w, 128, col0, acc);
#pragma unroll
    for (int i = 0; i < 8; ++i) {
      int m = i + 8 * khsel;
      xn[m * 128 + col0 + n] = acc[i] + node_b[col0 + n];
    }
  }
  __syncthreads();

  // stage 1b: h1 = leaky(t + t@cat1 + cat1_b + xn + xn@cat2 + cat2_b)
  {
    v8f acc = {};
    acc = wmma_k128(tA, cat1_w, 128, col0, acc);
    acc = wmma_k128(xn, cat2_w, 128, col0, acc);
#pragma unroll
    for (int i = 0; i < 8; ++i) {
      int m = i + 8 * khsel;
      float v = acc[i] + tA[m * 128 + col0 + n] + xn[m * 128 + col0 + n] +
                cat1_b[col0 + n] + cat2_b[col0 + n];
      h1[m * 128 + col0 + n] = leaky(v);
    }
  }
  __syncthreads();

  // stage 2: h2 = leaky(h1 @ W0t + b0)
  {
    v8f acc = {};
    acc = wmma_k128(h1, W0t, 128, col0, acc);
#pragma unroll
    for (int i = 0; i < 8; ++i) {
      int m = i + 8 * khsel;
      h2[m * 128 + col0 + n] = leaky(acc[i] + b0[col0 + n]);
    }
  }
  __syncthreads();

  // stage 3 (64 output cols -> waves 0..3): out = leaky(h2 @ W1t + b1)
  if (wave < 4) {
    v8f acc = {};
    acc = wmma_k128(h2, W1t, 64, col0, acc);
#pragma unroll
    for (int i = 0; i < 8; ++i) {
      int m = i + 8 * khsel;
      outp[(size_t)(tile0 + m) * 64 + col0 + n] = leaky(acc[i] + b1[col0 + n]);
    }
  }
}

extern "C" void kernel_launch(void* const* d_in, const int* in_sizes, int n_in,
                              void* d_out, int out_size, void* d_ws, size_t ws_size,
                              hipStream_t stream) {
  const float* x      = (const float*)d_in[0];
  const int*   ei     = (const int*)d_in[1];
  const float* ew     = (const float*)d_in[2];
  const float* wlat   = (const float*)d_in[3];
  const float* elw    = (const float*)d_in[4];
  const float* elb    = (const float*)d_in[5];
  const float* node_w = (const float*)d_in[6];
  const float* node_b = (const float*)d_in[7];
  const float* cat1_w = (const float*)d_in[8];
  const float* cat1_b = (const float*)d_in[9];
  const float* cat2_w = (const float*)d_in[10];
  const float* cat2_b = (const float*)d_in[11];
  const float* s0aw   = (const float*)d_in[12];
  const float* s0ab   = (const float*)d_in[13];
  const float* s0w    = (const float*)d_in[14];
  const float* s0b    = (const float*)d_in[15];
  const float* s1aw   = (const float*)d_in[16];
  const float* s1ab   = (const float*)d_in[17];
  const float* s1w    = (const float*)d_in[18];
  const float* s1b    = (const float*)d_in[19];
  float* out = (float*)d_out;

  float* ws      = (float*)d_ws;
  float* out_acc = ws;                                  // N*128
  float* styles0 = ws + (size_t)Nn * Hh;                // (H+H)*RANK = 2560
  float* styles1 = styles0 + (Hh + Hh) * RNK;           // (H+OUT)*RANK = 1920
  float* W0t     = styles1 + (Hh + OUTC) * RNK;         // 128*128
  float* W1t     = W0t + Hh * Hh;                       // 128*64

  zero_ws<<<2048, 256, 0, stream>>>(out_acc, Nn * Hh);
  styles_kernel<<<((Hh + Hh) * RNK + 255) / 256, 256, 0, stream>>>(
      wlat, s0aw, s0ab, styles0, (Hh + Hh) * RNK);
  styles_kernel<<<((Hh + OUTC) * RNK + 255) / 256, 256, 0, stream>>>(
      wlat + WDIM, s1aw, s1ab, styles1, (Hh + OUTC) * RNK);
  modw_kernel<<<Hh, Hh, 0, stream>>>(s0w, styles0, W0t, Hh, Hh);
  modw_kernel<<<OUTC, Hh, 0, stream>>>(s1w, styles1, W1t, OUTC, Hh);
  edge_kernel<<<(Ee + 1) / 2, 256, 0, stream>>>(elw, ei, ew, out_acc, Ee);
  fused_dense<<<Nn / 16, 256, 0, stream>>>(out_acc, elb, x, node_w, node_b,
                                           cat1_w, cat1_b, cat2_w, cat2_b,
                                           W0t, s0b, W1t, s1b, out);
}